// PositionEncoding_20667382628634
// MI455X (gfx1250) — compile-verified
//
#include <hip/hip_runtime.h>
#include <math.h>

#define B_    4
#define N_    8192
#define KNN   16
#define C1    64
#define H_    128
#define DIM_  256
#define BN_EPS   1e-5f
#define NORM_EPS 1e-12f
#define SLOPE    0.2f

typedef float v2f __attribute__((ext_vector_type(2)));
typedef float v8f __attribute__((ext_vector_type(8)));
typedef unsigned int u32x4 __attribute__((ext_vector_type(4)));
typedef int i32x4 __attribute__((ext_vector_type(4)));
typedef int i32x8 __attribute__((ext_vector_type(8)));

// ---- workspace layout (float offsets) ----
#define OFF_CENT   0            // 12: centroid [B][3]
#define OFF_GSUM   12           // scalar sum of g_dis
#define OFF_GSQ    13           // scalar sumsq of g_dis
#define OFF_ASUM   14           // scalar sum of alpha
#define OFF_ASQ    15           // scalar sumsq of alpha
#define OFF_DTS    16           // 64: dis branch BN1 scale per channel
#define OFF_DTO    80           // 64: dis branch BN1 offset
#define OFF_ATS    144          // 64: ang branch BN1 scale
#define OFF_ATO    208          // 64: ang branch BN1 offset
#define OFF_ZS_D   272          // 128: z sums (dis)
#define OFF_ZQ_D   400          // 128
#define OFF_ZS_A   528          // 128: z sums (ang)
#define OFF_ZQ_A   656          // 128
#define OFF_BNS_D  784          // 128: BN2 scale (dis)
#define OFF_BNO_D  912          // 128
#define OFF_BNS_A  1040         // 128: BN2 scale (ang)
#define OFF_BNO_A  1168         // 128
#define OFF_GDIS   1536                     // B*N
#define OFF_P2GCN  (OFF_GDIS + B_*N_)       // 3*B*N (normalized p2gc)
#define OFF_AMAX   (OFF_P2GCN + 3*B_*N_)    // B*N
#define OFF_AMIN   (OFF_AMAX + B_*N_)       // B*N

// ---------------------------------------------------------------------------
// Tensor Data Mover: 2-D tile load (f32 elements) into LDS.
// Descriptor packing per CDNA5 ISA (D# groups 0/1;
// groups 2/3 zero => 2-D tensor). data_size=2 (4 bytes), type=2 ("image"),
// count=1 valid user descriptor.
__device__ __forceinline__ void tdm_load_f32_2d(unsigned lds_addr, const void* gaddr,
                                                unsigned tensor_d0, unsigned tensor_d1,
                                                unsigned tile_d0, unsigned tile_d1,
                                                unsigned long long stride0) {
  unsigned long long ga = (unsigned long long)(uintptr_t)gaddr;
  u32x4 g0;
  g0.x = 1u;                                                   // count=1 (valid)
  g0.y = lds_addr;                                             // LDS byte address
  g0.z = (unsigned)(ga & 0xFFFFFFFFu);                         // global_addr[31:0]
  g0.w = (unsigned)((ga >> 32) & 0x01FFFFFFu) | (2u << 30);    // [56:32] + type=2
  i32x8 g1;
  g1[0] = (int)(2u << 16);                                     // data_size=4B
  g1[1] = (int)((tensor_d0 & 0xFFFFu) << 16);                  // tensor_dim0[15:0]
  g1[2] = (int)(((tensor_d0 >> 16) & 0xFFFFu) |
                ((tensor_d1 & 0xFFFFu) << 16));                // d0 hi | d1 lo
  g1[3] = (int)(((tensor_d1 >> 16) & 0xFFFFu) |
                ((tile_d0 & 0xFFFFu) << 16));                  // d1 hi | tile_dim0
  g1[4] = (int)(tile_d1 & 0xFFFFu);                            // tile_dim1 (dim2=0)
  g1[5] = (int)(unsigned)(stride0 & 0xFFFFFFFFu);              // dim0_stride[31:0]
  g1[6] = (int)(unsigned)((stride0 >> 32) & 0xFFFFu);          // dim0_stride[47:32]
  g1[7] = 0;
  i32x4 z4 = {0, 0, 0, 0};
#if defined(__clang_major__) && (__clang_major__ >= 23)
  i32x8 z8 = {0, 0, 0, 0, 0, 0, 0, 0};
  __builtin_amdgcn_tensor_load_to_lds(g0, g1, z4, z4, z8, 0);
#else
  __builtin_amdgcn_tensor_load_to_lds(g0, g1, z4, z4, 0);
#endif
}

// ---------------------------------------------------------------------------
__global__ __launch_bounds__(256) void k_init(float* __restrict__ ws) {
  int i = blockIdx.x * 256 + threadIdx.x;
  if (i < 1536) ws[i] = 0.f;
}

// ---------------------------------------------------------------------------
__global__ __launch_bounds__(256) void k_centroid(const float* __restrict__ pts,
                                                  float* __restrict__ ws) {
  __shared__ float red[256];
  const int b = blockIdx.x;
  for (int c = 0; c < 3; ++c) {
    float s = 0.f;
    for (int n = threadIdx.x; n < N_; n += 256)
      s += pts[(size_t)b*3*N_ + (size_t)c*N_ + n];
    red[threadIdx.x] = s;
    __syncthreads();
    for (int off = 128; off > 0; off >>= 1) {
      if (threadIdx.x < off) red[threadIdx.x] += red[threadIdx.x + off];
      __syncthreads();
    }
    if (threadIdx.x == 0) ws[OFF_CENT + b*3 + c] = red[0] * (1.f / N_);
    __syncthreads();
  }
}

// g_dis + normalized p2gc + global scalar stats of g_dis
__global__ __launch_bounds__(256) void k_gdis(const float* __restrict__ pts,
                                              float* __restrict__ ws) {
  __shared__ float r1[256], r2[256];
  const int i = blockIdx.x * 256 + threadIdx.x;
  const int b = i >> 13, n = i & (N_ - 1);
  const float cx = ws[OFF_CENT + b*3 + 0];
  const float cy = ws[OFF_CENT + b*3 + 1];
  const float cz = ws[OFF_CENT + b*3 + 2];
  const float* P = pts + (size_t)b*3*N_;
  float x = P[n] - cx, y = P[N_ + n] - cy, z = P[2*N_ + n] - cz;
  float g = x*x + y*y + z*z;
  ws[OFF_GDIS + i] = g;
  float inv = 1.f / fmaxf(sqrtf(g), NORM_EPS);
  float* G = ws + OFF_P2GCN + (size_t)b*3*N_;
  G[n] = x*inv; G[N_ + n] = y*inv; G[2*N_ + n] = z*inv;
  r1[threadIdx.x] = g; r2[threadIdx.x] = g*g;
  __syncthreads();
  for (int off = 128; off > 0; off >>= 1) {
    if (threadIdx.x < off) { r1[threadIdx.x] += r1[threadIdx.x+off]; r2[threadIdx.x] += r2[threadIdx.x+off]; }
    __syncthreads();
  }
  if (threadIdx.x == 0) { atomicAdd(&ws[OFF_GSUM], r1[0]); atomicAdd(&ws[OFF_GSQ], r2[0]); }
}

// ---------------------------------------------------------------------------
// kNN via WMMA f32 16x16x4 + fused top-16 selection + alpha extremes.
// Column points are staged into LDS by the Tensor Data Mover (double-buffered,
// one 3x128 f32 tile per stage), WMMA consumes them; 16 threads per row keep
// private top-16 lists, merged per row at the end.
__global__ __launch_bounds__(256) void k_knn(const float* __restrict__ pts,
                                             float* __restrict__ ws) {
  const int tid  = threadIdx.x;
  const int lane = tid & 31;
  const int wave = tid >> 5;
  const int hi   = lane >> 4;    // 0 | 1
  const int l    = lane & 15;

  const int b  = blockIdx.x / (N_/16);
  const int m0 = (blockIdx.x % (N_/16)) * 16;
  const float* P = pts + (size_t)b*3*N_;

  __shared__ float colb[2][3*128];     // TDM landing buffers (3 KB)
  __shared__ float xxr[16];
  __shared__ float dist[16][128];
  __shared__ float mval[256*16];
  __shared__ int   mid [256*16];
  __shared__ int   idxb[16][16];
  __shared__ float ared[256];
  __shared__ float ared2[256];

  // prime TDM pipeline: stage 0 columns -> colb[0]
  if (wave == 0) {
    tdm_load_f32_2d((unsigned)(uintptr_t)&colb[0][0], P, N_, 3, 128, 3, N_);
  }

  // A fragment: rows m0..m0+15, K = {x,y,z,0}
  float px = P[0*N_ + m0 + l];
  float py = P[1*N_ + m0 + l];
  float pz = P[2*N_ + m0 + l];
  v2f afrag;
  afrag.x = hi ? pz : px;
  afrag.y = hi ? 0.f : py;
  if (wave == 0 && lane < 16) xxr[l] = px*px + py*py + pz*pz;

  float tv[16]; int tix[16];
  #pragma unroll
  for (int t = 0; t < 16; ++t) { tv[t] = -INFINITY; tix[t] = 0; }
  const int r_sel = tid >> 4;
  const int s_sel = tid & 15;

  for (int stage = 0; stage < 64; ++stage) {
    if (wave == 0) {
      if (stage + 1 < 64) {
        tdm_load_f32_2d((unsigned)(uintptr_t)&colb[(stage+1)&1][0],
                        P + (size_t)(stage+1)*128, N_, 3, 128, 3, N_);
        __builtin_amdgcn_s_wait_tensorcnt(1);   // current stage's tile landed
      } else {
        __builtin_amdgcn_s_wait_tensorcnt(0);
      }
    }
    __syncthreads();   // colb[stage&1] visible to all waves

    // compute phase: this wave's 16x16 tile of neg squared distances
    const float* cb = &colb[stage & 1][0];
    const int colL = wave*16 + l;
    float cx = cb[0*128 + colL];
    float cy = cb[1*128 + colL];
    float cz = cb[2*128 + colL];
    v2f bfrag;
    bfrag.x = hi ? cz : cx;
    bfrag.y = hi ? 0.f : cy;
    v8f c8 = {};
    c8 = __builtin_amdgcn_wmma_f32_16x16x4_f32(false, afrag, false, bfrag,
                                               (short)0, c8, false, false);
    float xxc = cx*cx + cy*cy + cz*cz;
    #pragma unroll
    for (int r = 0; r < 8; ++r) {
      int rr = hi*8 + r;
      dist[rr][colL] = 2.f*c8[r] - xxr[rr] - xxc;   // neg_d (<= 0, self == 0)
    }
    __syncthreads();

    // selection phase: thread (r_sel, s_sel) scans its 8 stage-local columns
    const int basecol = stage * 128;
    #pragma unroll
    for (int q = 0; q < 8; ++q) {
      int clc = s_sel*8 + q;
      float x = dist[r_sel][clc];
      if (x > tv[0]) {               // tv ascending, tv[0] = current min of top-16
        tv[0] = x; tix[0] = basecol + clc;
        #pragma unroll
        for (int p = 0; p < 15; ++p) {
          if (tv[p] > tv[p+1]) {
            float tf = tv[p];  tv[p]  = tv[p+1];  tv[p+1]  = tf;
            int   ti = tix[p]; tix[p] = tix[p+1]; tix[p+1] = ti;
          }
        }
      }
    }
    __syncthreads();
  }

  // merge: 16 lists of 16 per row -> top-16 of the row
  #pragma unroll
  for (int t = 0; t < 16; ++t) { mval[tid*16 + t] = tv[t]; mid[tid*16 + t] = tix[t]; }
  __syncthreads();
  if (s_sel == 0) {
    const int base = r_sel * 256;
    for (int t = 0; t < 16; ++t) {
      float best = -INFINITY; int bi = 0;
      for (int q = 0; q < 256; ++q) {
        float v = mval[base + q];
        if (v > best) { best = v; bi = q; }
      }
      idxb[r_sel][t] = mid[base + bi];
      mval[base + bi] = -INFINITY;
    }
  }
  __syncthreads();

  // alpha for (row r_sel, neighbor s_sel)
  const int n  = m0 + r_sel;
  const int nb = idxb[r_sel][s_sel];
  float qx = P[0*N_ + n], qy = P[1*N_ + n], qz = P[2*N_ + n];
  float ex = P[0*N_ + nb] - qx, ey = P[1*N_ + nb] - qy, ez = P[2*N_ + nb] - qz;
  float inv = 1.f / fmaxf(sqrtf(ex*ex + ey*ey + ez*ez), NORM_EPS);
  const float* G = ws + OFF_P2GCN + (size_t)b*3*N_;
  float al = (ex*G[0*N_ + n] + ey*G[1*N_ + n] + ez*G[2*N_ + n]) * inv;

  ared[tid] = al;
  __syncthreads();
  if (s_sel == 0) {
    float mx = -INFINITY, mn = INFINITY;
    for (int j = 0; j < 16; ++j) {
      float v = ared[r_sel*16 + j];
      mx = fmaxf(mx, v); mn = fminf(mn, v);
    }
    ws[OFF_AMAX + (size_t)b*N_ + n] = mx;
    ws[OFF_AMIN + (size_t)b*N_ + n] = mn;
  }
  __syncthreads();
  ared2[tid] = al * al;
  __syncthreads();
  for (int off = 128; off > 0; off >>= 1) {
    if (tid < off) { ared[tid] += ared[tid+off]; ared2[tid] += ared2[tid+off]; }
    __syncthreads();
  }
  if (tid == 0) { atomicAdd(&ws[OFF_ASUM], ared[0]); atomicAdd(&ws[OFF_ASQ], ared2[0]); }
}

// ---------------------------------------------------------------------------
// BN1 channel tables (analytic: x = W1[c] * scalar)
__global__ void k_tab1(float* __restrict__ ws,
                       const float* __restrict__ Wd1, const float* __restrict__ gd1,
                       const float* __restrict__ bd1,
                       const float* __restrict__ Wa1, const float* __restrict__ ga1,
                       const float* __restrict__ ba1) {
  int c = threadIdx.x;
  if (c >= C1) return;
  const float cg = (float)(B_*N_);
  const float ca = (float)(B_*N_*KNN);
  float mg = ws[OFF_GSUM] / cg;
  float vg = ws[OFF_GSQ] / cg - mg*mg;
  float ma = ws[OFF_ASUM] / ca;
  float va = ws[OFF_ASQ] / ca - ma*ma;

  float w  = Wd1[c];
  float r  = rsqrtf(w*w*vg + BN_EPS);
  float sc = w * r * gd1[c];
  ws[OFF_DTS + c] = sc;
  ws[OFF_DTO + c] = bd1[c] - mg * sc;

  float wa = Wa1[c];
  float ra = rsqrtf(wa*wa*va + BN_EPS);
  float sa = wa * ra * ga1[c];
  ws[OFF_ATS + c] = sa;
  ws[OFF_ATO + c] = ba1[c] - ma * sa;
}

// ---------------------------------------------------------------------------
// Feature GEMM via WMMA f32 16x16x4: z = W2[128x64] * d[64 x pts].
// W2 staged into LDS by the TDM while the activation tile is built.
// d[c] per point is lrelu(s_c * x + o_c) where x = (s_c>=0 ? xpos : xneg).
// finalize==0: accumulate per-channel sum/sumsq.  finalize==1: BN2+lrelu -> out.
__global__ __launch_bounds__(256) void k_gemm(const float* __restrict__ W2,
                                              const float* __restrict__ tabS,
                                              const float* __restrict__ tabO,
                                              const float* __restrict__ xpos,
                                              const float* __restrict__ xneg,
                                              float* __restrict__ zsum,
                                              float* __restrict__ zsq,
                                              const float* __restrict__ bnS,
                                              const float* __restrict__ bnO,
                                              float* __restrict__ out,
                                              int chOff, int finalize) {
  __shared__ float w2ls[H_*C1];     // 32 KB (TDM destination)
  __shared__ float dls[C1][64];     // 16 KB
  const int tid  = threadIdx.x;
  const int lane = tid & 31;
  const int wave = tid >> 5;
  const int hi   = lane >> 4;
  const int l    = lane & 15;
  const int pt0  = blockIdx.x * 64;

  // async: DMA the whole 128x64 weight block into LDS (1-D tile)
  if (wave == 0) {
    tdm_load_f32_2d((unsigned)(uintptr_t)&w2ls[0], W2, H_*C1, 1, H_*C1, 1, H_*C1);
  }
  __builtin_prefetch(&xpos[pt0 + (tid & 63)], 0, 3);

  // build activation tile while the weights are in flight
  for (int e = tid; e < C1*64; e += 256) {
    int c = e >> 6, p = e & 63;
    float s = tabS[c];
    float x = (s >= 0.f) ? xpos[pt0 + p] : xneg[pt0 + p];
    float y = fmaf(s, x, tabO[c]);
    dls[c][p] = (y >= 0.f) ? y : SLOPE * y;
  }
  if (wave == 0) __builtin_amdgcn_s_wait_tensorcnt(0);
  __syncthreads();

  const int m0 = wave * 16;     // 8 waves cover M = 0..127
  for (int nt = 0; nt < 4; ++nt) {
    const int n0 = nt * 16;
    v8f acc = {};
    #pragma unroll
    for (int k0 = 0; k0 < C1; k0 += 4) {
      v2f A, Bf;
      A.x  = w2ls[(m0 + l)*C1 + k0 + (hi ? 2 : 0)];
      A.y  = w2ls[(m0 + l)*C1 + k0 + (hi ? 3 : 1)];
      Bf.x = dls[k0 + (hi ? 2 : 0)][n0 + l];
      Bf.y = dls[k0 + (hi ? 3 : 1)][n0 + l];
      acc = __builtin_amdgcn_wmma_f32_16x16x4_f32(false, A, false, Bf,
                                                  (short)0, acc, false, false);
    }
    if (!finalize) {
      #pragma unroll
      for (int r = 0; r < 8; ++r) {
        float v = acc[r]; float q = v*v;
        v += __shfl_xor(v, 1);  q += __shfl_xor(q, 1);
        v += __shfl_xor(v, 2);  q += __shfl_xor(q, 2);
        v += __shfl_xor(v, 4);  q += __shfl_xor(q, 4);
        v += __shfl_xor(v, 8);  q += __shfl_xor(q, 8);
        if (l == 0) {
          int M = m0 + r + (hi ? 8 : 0);
          atomicAdd(&zsum[M], v);
          atomicAdd(&zsq[M], q);
        }
      }
    } else {
      #pragma unroll
      for (int r = 0; r < 8; ++r) {
        int M  = m0 + r + (hi ? 8 : 0);
        int gp = pt0 + n0 + l;
        int b  = gp >> 13, nn = gp & (N_ - 1);
        float y = fmaf(acc[r], bnS[M], bnO[M]);
        y = (y >= 0.f) ? y : SLOPE * y;
        out[((size_t)b*DIM_ + chOff + M)*N_ + nn] = y;
      }
    }
  }
}

// ---------------------------------------------------------------------------
__global__ void k_tab2(float* __restrict__ ws,
                       const float* __restrict__ gd2, const float* __restrict__ bd2,
                       const float* __restrict__ ga2, const float* __restrict__ ba2) {
  int t = threadIdx.x;
  if (t >= 256) return;
  int M = t & 127;
  const float cnt = (float)(B_*N_);
  if (t < 128) {
    float mean = ws[OFF_ZS_D + M] / cnt;
    float var  = ws[OFF_ZQ_D + M] / cnt - mean*mean;
    float r    = rsqrtf(var + BN_EPS);
    float sc   = r * gd2[M];
    ws[OFF_BNS_D + M] = sc;
    ws[OFF_BNO_D + M] = bd2[M] - mean * sc;
  } else {
    float mean = ws[OFF_ZS_A + M] / cnt;
    float var  = ws[OFF_ZQ_A + M] / cnt - mean*mean;
    float r    = rsqrtf(var + BN_EPS);
    float sc   = r * ga2[M];
    ws[OFF_BNS_A + M] = sc;
    ws[OFF_BNO_A + M] = ba2[M] - mean * sc;
  }
}

// ---------------------------------------------------------------------------
extern "C" void kernel_launch(void* const* d_in, const int* in_sizes, int n_in,
                              void* d_out, int out_size, void* d_ws, size_t ws_size,
                              hipStream_t stream) {
  (void)in_sizes; (void)n_in; (void)out_size; (void)ws_size;
  const float* pts = (const float*)d_in[0];
  // d_in[1] is k (==16), fixed at compile time
  const float* Wd1 = (const float*)d_in[2];
  const float* gd1 = (const float*)d_in[3];
  const float* bd1 = (const float*)d_in[4];
  const float* Wd2 = (const float*)d_in[5];
  const float* gd2 = (const float*)d_in[6];
  const float* bd2 = (const float*)d_in[7];
  const float* Wa1 = (const float*)d_in[8];
  const float* ga1 = (const float*)d_in[9];
  const float* ba1 = (const float*)d_in[10];
  const float* Wa2 = (const float*)d_in[11];
  const float* ga2 = (const float*)d_in[12];
  const float* ba2 = (const float*)d_in[13];
  float* out = (float*)d_out;
  float* ws  = (float*)d_ws;

  k_init    <<<6, 256, 0, stream>>>(ws);
  k_centroid<<<B_, 256, 0, stream>>>(pts, ws);
  k_gdis    <<<B_*N_/256, 256, 0, stream>>>(pts, ws);
  k_knn     <<<B_*(N_/16), 256, 0, stream>>>(pts, ws);
  k_tab1    <<<1, 64, 0, stream>>>(ws, Wd1, gd1, bd1, Wa1, ga1, ba1);

  const int gemmGrid = B_*N_/64;
  // stats passes
  k_gemm<<<gemmGrid, 256, 0, stream>>>(Wd2, ws+OFF_DTS, ws+OFF_DTO,
                                       ws+OFF_GDIS, ws+OFF_GDIS,
                                       ws+OFF_ZS_D, ws+OFF_ZQ_D,
                                       nullptr, nullptr, out, 0, 0);
  k_gemm<<<gemmGrid, 256, 0, stream>>>(Wa2, ws+OFF_ATS, ws+OFF_ATO,
                                       ws+OFF_AMAX, ws+OFF_AMIN,
                                       ws+OFF_ZS_A, ws+OFF_ZQ_A,
                                       nullptr, nullptr, out, 128, 0);
  k_tab2<<<1, 256, 0, stream>>>(ws, gd2, bd2, ga2, ba2);
  // finalize passes
  k_gemm<<<gemmGrid, 256, 0, stream>>>(Wd2, ws+OFF_DTS, ws+OFF_DTO,
                                       ws+OFF_GDIS, ws+OFF_GDIS,
                                       ws+OFF_ZS_D, ws+OFF_ZQ_D,
                                       ws+OFF_BNS_D, ws+OFF_BNO_D, out, 0, 1);
  k_gemm<<<gemmGrid, 256, 0, stream>>>(Wa2, ws+OFF_ATS, ws+OFF_ATO,
                                       ws+OFF_AMAX, ws+OFF_AMIN,
                                       ws+OFF_ZS_A, ws+OFF_ZQ_A,
                                       ws+OFF_BNS_A, ws+OFF_BNO_A, out, 128, 1);
}